// ConnectivityLoss_29549374996914
// MI455X (gfx1250) — compile-verified
//
#include <hip/hip_runtime.h>
#include <hip/hip_bf16.h>
#include <stdint.h>

// ---------------- problem constants ----------------
#define B_   8u
#define N_   2097152u            // 128*128*128 = 2^21
#define TOT_ (B_ * N_)           // 16,777,216 voxels
#define W_   128u
#define HW_  16384u              // 128*128

// ---------------- helpers ----------------
__device__ __forceinline__ bool mbit(const unsigned* __restrict__ mw, unsigned i) {
    return (mw[i >> 5] >> (i & 31u)) & 1u;
}

__device__ __forceinline__ int findRoot(volatile int* p, int i) {
    int r = p[i];
    while (true) {
        int q = p[r];
        if (q == r) return r;
        r = q;
    }
}

__device__ __forceinline__ void unite(int* p, int a, int b) {
    a = findRoot(p, a);
    b = findRoot(p, b);
    while (a != b) {
        if (a > b) { int t = a; a = b; b = t; }     // a < b
        int old = atomicMin(&p[b], a);
        if (old == b) return;                       // successfully linked b -> a
        b = findRoot(p, old);                       // someone else linked; continue
        a = findRoot(p, a);
    }
}

// ---------------- kernel 0: zero counts + stats (int4 stores) ----------------
extern "C" __global__ void __launch_bounds__(256) cc_zero(int4* __restrict__ counts4) {
    unsigned i = blockIdx.x * 256u + threadIdx.x;
    if (i < (TOT_ + 64u) / 4u) counts4[i] = make_int4(0, 0, 0, 0);
}

// ---------------- kernel 1: mask pack (ballot) + parent init ----------------
// Stages 2048 floats/block into LDS via the CDNA5 async-to-LDS path
// (two global_load_async_to_lds_b128 in flight per lane, tracked by ASYNCcnt),
// then each wave ballots the >0.5 predicate into 32-bit mask words.
extern "C" __global__ void __launch_bounds__(256) cc_pack_init(
        const float* __restrict__ vox,
        int* __restrict__ parent,
        unsigned* __restrict__ maskw) {
    __shared__ __align__(16) float stage[2048];

    const unsigned tid  = threadIdx.x;
    const unsigned base = blockIdx.x * 2048u;           // first voxel of block

    // per-lane LDS byte address (low 32 bits of the generic shared pointer)
    unsigned ldsAddr0 = (unsigned)(uintptr_t)(&stage[0]) + tid * 16u;
    unsigned ldsAddr1 = ldsAddr0 + 4096u;
    // uniform global base for this block (SGPR pair), per-lane 32-bit offsets
    unsigned long long gbase = (unsigned long long)(uintptr_t)(vox + base);
    unsigned voff0 = tid * 16u;
    unsigned voff1 = voff0 + 4096u;

    asm volatile("global_load_async_to_lds_b128 %0, %1, %2"
                 :: "v"(ldsAddr0), "v"(voff0), "s"(gbase) : "memory");
    asm volatile("global_load_async_to_lds_b128 %0, %1, %2"
                 :: "v"(ldsAddr1), "v"(voff1), "s"(gbase) : "memory");
    asm volatile("s_wait_asynccnt 0" ::: "memory");
    __syncthreads();

    const unsigned lane = tid & 31u;
    const unsigned wave = tid >> 5;
#pragma unroll
    for (int k = 0; k < 8; ++k) {
        float f = stage[k * 256 + tid];
        bool  pred = f > 0.5f;
        unsigned w = __builtin_amdgcn_ballot_w32(pred);
        if (lane == 0u) maskw[(base >> 5) + (unsigned)k * 8u + wave] = w;
    }
#pragma unroll
    for (int k = 0; k < 8; ++k) {
        unsigned i = base + (unsigned)k * 256u + tid;
        parent[i] = (int)i;
    }
}

// ---------------- kernel 2: union-find merge over +x,+y,+z edges ----------------
extern "C" __global__ void __launch_bounds__(256) cc_merge(
        int* __restrict__ parent, const unsigned* __restrict__ maskw) {
    unsigned i = blockIdx.x * 256u + threadIdx.x;
    if (!mbit(maskw, i)) return;
    unsigned r = i & (N_ - 1u);
    unsigned x = r & (W_ - 1u);
    unsigned y = (r >> 7) & (W_ - 1u);
    unsigned z = r >> 14;
    if (x < W_ - 1u && mbit(maskw, i + 1u))   unite(parent, (int)i, (int)(i + 1u));
    if (y < W_ - 1u && mbit(maskw, i + W_))   unite(parent, (int)i, (int)(i + W_));
    if (z < W_ - 1u && mbit(maskw, i + HW_))  unite(parent, (int)i, (int)(i + HW_));
}

// ---------------- kernel 3: path compression ----------------
extern "C" __global__ void __launch_bounds__(256) cc_flatten(
        int* __restrict__ parent, const unsigned* __restrict__ maskw) {
    unsigned i = blockIdx.x * 256u + threadIdx.x;
    if (!mbit(maskw, i)) return;
    parent[i] = findRoot(parent, (int)i);
}

// ---------------- kernel 4: component-size histogram via LDS hash ----------------
#define HASHSZ 2048u

__device__ __forceinline__ void hinsert(int* hkey, int* hval, int key, int val,
                                        int* __restrict__ counts) {
    unsigned h = ((unsigned)key * 2654435761u) >> 21u;   // 11-bit slot
    for (int probe = 0; probe < 32; ++probe) {
        int k = hkey[h];
        if (k == key) { atomicAdd(&hval[h], val); return; }
        if (k == -1) {
            int old = atomicCAS(&hkey[h], -1, key);
            if (old == -1 || old == key) { atomicAdd(&hval[h], val); return; }
        }
        h = (h + 1u) & (HASHSZ - 1u);
    }
    atomicAdd(&counts[key], val);                        // hash full: direct
}

// One mask word (32 consecutive voxels) per thread; run-aggregate equal roots,
// merge in LDS hash, flush a few global atomics per block. This collapses the
// ~8M same-root additions of the giant component (p=0.5 >> percolation
// threshold) into O(blocks) global atomics.
extern "C" __global__ void __launch_bounds__(256) cc_count(
        const int* __restrict__ parent,
        const unsigned* __restrict__ maskw,
        int* __restrict__ counts) {
    __shared__ int hkey[HASHSZ];
    __shared__ int hval[HASHSZ];
    for (unsigned s = threadIdx.x; s < HASHSZ; s += 256u) { hkey[s] = -1; hval[s] = 0; }
    __syncthreads();

    unsigned wi = blockIdx.x * 256u + threadIdx.x;       // mask-word index
    unsigned w  = maskw[wi];
    unsigned i0 = wi * 32u;
    int cur = -1, acc = 0;
    while (w) {
        unsigned j = (unsigned)__builtin_ctz(w);
        w &= w - 1u;
        int r = parent[i0 + j];                          // already flattened
        if (r == cur) { acc++; }
        else {
            if (acc) hinsert(hkey, hval, cur, acc, counts);
            cur = r; acc = 1;
        }
    }
    if (acc) hinsert(hkey, hval, cur, acc, counts);
    __syncthreads();

    for (unsigned s = threadIdx.x; s < HASHSZ; s += 256u) {
        int k = hkey[s], v = hval[s];
        if (k >= 0 && v > 0) atomicAdd(&counts[k], v);
    }
}

// ---------------- kernel 5: per-sample stats (int4 scan) ----------------
// stats layout: total[0..7], largest[8..15], ncomp[16..23]
extern "C" __global__ void __launch_bounds__(256) cc_stats(
        const int4* __restrict__ counts4, int* __restrict__ stats) {
    unsigned i4 = blockIdx.x * 256u + threadIdx.x;       // index into int4 view
    int4 c4 = counts4[i4];
    int c[4] = { c4.x, c4.y, c4.z, c4.w };
#pragma unroll
    for (unsigned k = 0; k < 4; ++k) {
        if (c[k] > 0) {                                  // root of a component
            unsigned b = (i4 * 4u + k) >> 21;            // voxel / N_
            atomicAdd(&stats[b], c[k]);                  // total foreground
            atomicMax(&stats[8u + b], c[k]);             // largest component
            atomicAdd(&stats[16u + b], 1);               // #components
        }
    }
}

// ---------------- kernel 6: final scalar ----------------
extern "C" __global__ void cc_final(const int* __restrict__ stats,
                                    float* __restrict__ out) {
    if (threadIdx.x == 0 && blockIdx.x == 0) {
        float s = 0.0f;
        for (unsigned b = 0; b < B_; ++b) {
            float total   = (float)stats[b];
            float largest = (float)stats[8u + b];
            int   ncomp   = stats[16u + b];
            float frac = (total - largest) / (total + 1e-6f);
            s += (ncomp > 1) ? frac : 0.0f;
        }
        out[0] = 10.0f * s / (float)B_;                  // PENALTY = 10
    }
}

// ---------------- host launcher ----------------
extern "C" void kernel_launch(void* const* d_in, const int* in_sizes, int n_in,
                              void* d_out, int out_size, void* d_ws, size_t ws_size,
                              hipStream_t stream) {
    (void)in_sizes; (void)n_in; (void)out_size; (void)ws_size;
    const float* vox = (const float*)d_in[0];
    float* out = (float*)d_out;

    // workspace layout (bytes):
    //   parent : [0,            64 MiB)
    //   counts : [64 MiB,      128 MiB)
    //   stats  : [128 MiB,     128 MiB + 256 B)   (64 ints; contiguous w/ counts)
    //   maskw  : [128 MiB+256, +2 MiB)
    char* w = (char*)d_ws;
    int*      parent = (int*)w;
    int*      counts = (int*)(w + (size_t)TOT_ * 4u);
    int*      stats  = counts + TOT_;                    // 64 ints after counts
    unsigned* maskw  = (unsigned*)(w + (size_t)TOT_ * 8u + 256u);

    cc_zero     <<<((TOT_ + 64u) / 4u + 255u) / 256u, 256, 0, stream>>>((int4*)counts);
    cc_pack_init<<<TOT_ / 2048u,                      256, 0, stream>>>(vox, parent, maskw);
    cc_merge    <<<TOT_ / 256u,                       256, 0, stream>>>(parent, maskw);
    cc_flatten  <<<TOT_ / 256u,                       256, 0, stream>>>(parent, maskw);
    cc_count    <<<TOT_ / 8192u,                      256, 0, stream>>>(parent, maskw, counts);
    cc_stats    <<<TOT_ / 1024u,                      256, 0, stream>>>((const int4*)counts, stats);
    cc_final    <<<1,                                 32,  0, stream>>>(stats, out);
}